// SpaceLoss_88656714924201
// MI455X (gfx1250) — compile-verified
//
#include <hip/hip_runtime.h>
#include <hip/hip_bf16.h>

// ---------------------------------------------------------------------------
// SpaceLoss (proxy triplet) for MI455X / gfx1250, wave32.
//   B=4096, D=1024, N_IDS=100000, K=8192, MARGIN=0.03
// Dominant work: fp32 GEMM 4096x8192x1024 (68.7 GFLOP) fused with per-row
// top-2 reduction -> V_WMMA_F32_16X16X4_F32 (exact fp32 matrix math).
// A tile (16x1024 f32) staged in LDS with a per-row ROTATION swizzle plus a
// 64-float wrap replica (row stride 1088): reads are affine in d, so every
// ds_load_b64 uses an immediate offset (no per-load VALU), while remaining
// bank-conflict-free (bank = (4*ln + d + kh) mod 64 covers all 64 banks).
// ---------------------------------------------------------------------------

typedef float v2f __attribute__((ext_vector_type(2)));
typedef float v4f __attribute__((ext_vector_type(4)));
typedef float v8f __attribute__((ext_vector_type(8)));

#define SL_B 4096
#define SL_D 1024
#define SL_K 8192
#define SL_MARGIN 0.03f
#define NEG_INF_F (-3.402823466e38f)
#define SL_LDS_STRIDE 1088   // 1024 + 64-float wrap replica, mult of 64

static __device__ __forceinline__ v2f ldv2(const float* __restrict__ p) {
    return *reinterpret_cast<const v2f*>(p);
}
static __device__ __forceinline__ v4f ldv4(const float* __restrict__ p) {
    return *reinterpret_cast<const v4f*>(p);
}

static __device__ __forceinline__ v8f wmma_f32_k4(v2f a, v2f b, v8f c) {
    // D = A(16x4,f32) * B(4x16,f32) + C(16x16,f32)
    return __builtin_amdgcn_wmma_f32_16x16x4_f32(
        /*neg_a=*/false, a, /*neg_b=*/false, b,
        /*c_mod=*/(short)0, c, /*reuse_a=*/false, /*reuse_b=*/false);
}

// ---------------------------------------------------------------------------
// Kernel 1: ap[b] = 1 - <ftr[b], proto[label[b]]>. One wave per row.
// ---------------------------------------------------------------------------
__global__ __launch_bounds__(256) void sl_ap_kernel(
    const float* __restrict__ ftr, const float* __restrict__ protos,
    const int* __restrict__ label, float* __restrict__ ap) {
    const int gw   = (blockIdx.x * blockDim.x + threadIdx.x) >> 5;  // row b
    const int lane = threadIdx.x & 31;
    if (gw >= SL_B) return;

    const float* __restrict__ fr = ftr + (size_t)gw * SL_D;
    const float* __restrict__ pr = protos + (size_t)label[gw] * SL_D;

    float sum = 0.0f;
#pragma unroll
    for (int it = 0; it < SL_D / 128; ++it) {
        const int j = it * 128 + lane * 4;
        v4f a = ldv4(fr + j);
        v4f p = ldv4(pr + j);
        sum = fmaf(a.x, p.x, fmaf(a.y, p.y, fmaf(a.z, p.z, fmaf(a.w, p.w, sum))));
    }
#pragma unroll
    for (int m = 16; m >= 1; m >>= 1) sum += __shfl_xor(sum, m, 32);
    if (lane == 0) ap[gw] = 1.0f - sum;
}

// ---------------------------------------------------------------------------
// Kernel 2: fused GEMM + per-row top-2 smallest distance + argmin + loss.
// One block (8 waves, 256 thr) owns 16 rows of B across all K columns.
// A tile in LDS (rotation-swizzled, affine reads); B streamed from L2.
// ---------------------------------------------------------------------------
__global__ __launch_bounds__(256) void sl_an_topk_kernel(
    const float* __restrict__ ftr, const float* __restrict__ protos,
    const int* __restrict__ label, const int* __restrict__ lbl,
    const float* __restrict__ ap, float* __restrict__ row_loss) {
    extern __shared__ float sA[];        // 16 rows x SL_LDS_STRIDE floats
    const int lane = threadIdx.x & 31;
    const int w    = threadIdx.x >> 5;   // wave 0..7
    const int half = lane >> 4;          // 0: lanes 0-15, 1: lanes 16-31
    const int ln   = lane & 15;
    const int r0   = blockIdx.x * 16;    // row tile base

    // ---- Stage A tile. Row r is rotated left by 4*r floats:
    //   phys_col = (col + 4*r) mod 1024.
    // Coalesced b128 global loads, b128 LDS stores.
#pragma unroll
    for (int t = threadIdx.x; t < 16 * (SL_D / 4); t += 256) {
        const int row  = t >> 8;                    // 256 v4f per row
        const int col4 = (t & 255) * 4;
        const int scol = (col4 + row * 4) & (SL_D - 1);
        *reinterpret_cast<v4f*>(sA + row * SL_LDS_STRIDE + scol) =
            ldv4(ftr + (size_t)(r0 + row) * SL_D + col4);
    }
    __syncthreads();
    // Replicate the first 64 physical floats of each row at its tail so the
    // rotated reads (max offset 1082) never need a mod: 16 rows x 16 v4f.
    {
        const int row = threadIdx.x >> 4;           // 16 v4f per row
        const int q   = (threadIdx.x & 15) * 4;
        *reinterpret_cast<v4f*>(sA + row * SL_LDS_STRIDE + SL_D + q) =
            *reinterpret_cast<const v4f*>(sA + row * SL_LDS_STRIDE + q);
    }
    __syncthreads();

    const int kh = 2 * half;                        // k-offset of this half
    // Lane-constant base: logical col c of row ln lives at base + c.
    const float* __restrict__ aBase = sA + ln * SL_LDS_STRIDE + ln * 4 + kh;

    // Running per-lane top-2 (largest dot == smallest distance) for the 8
    // rows this half-wave carries in the C layout (m = i + 8*half).
    float best1[8], best2[8];
    int   idx1[8];
#pragma unroll
    for (int i = 0; i < 8; ++i) { best1[i] = NEG_INF_F; best2[i] = NEG_INF_F; idx1[i] = 0; }

    for (int ct = w; ct < SL_K / 16; ct += 8) {          // wave-uniform trips
        const int c0   = ct * 16;
        const int kcol = c0 + ln;                        // this lane's column n
        const int pid  = lbl[kcol];
        const float* __restrict__ bptr = protos + (size_t)pid * SL_D + kh;

        // 4 independent accumulator chains hide WMMA->WMMA RAW latency.
        v8f acc0 = {0.f,0.f,0.f,0.f,0.f,0.f,0.f,0.f};
        v8f acc1 = {0.f,0.f,0.f,0.f,0.f,0.f,0.f,0.f};
        v8f acc2 = {0.f,0.f,0.f,0.f,0.f,0.f,0.f,0.f};
        v8f acc3 = {0.f,0.f,0.f,0.f,0.f,0.f,0.f,0.f};
#pragma unroll 4
        for (int d = 0; d < SL_D; d += 16) {
            // Affine addresses -> ds_load_b64 / global_load_b64 with
            // immediate offsets; no per-load address VALU.
            acc0 = wmma_f32_k4(ldv2(aBase + d +  0), ldv2(bptr + d +  0), acc0);
            acc1 = wmma_f32_k4(ldv2(aBase + d +  4), ldv2(bptr + d +  4), acc1);
            acc2 = wmma_f32_k4(ldv2(aBase + d +  8), ldv2(bptr + d +  8), acc2);
            acc3 = wmma_f32_k4(ldv2(aBase + d + 12), ldv2(bptr + d + 12), acc3);
        }

        // Fold this tile's column (n = ln) into per-row running top-2.
#pragma unroll
        for (int i = 0; i < 8; ++i) {
            const float v = acc0[i] + acc1[i] + acc2[i] + acc3[i];  // dot
            const bool gt  = v > best1[i];
            const float nb2 = gt ? best1[i] : (v > best2[i] ? v : best2[i]);
            idx1[i]  = gt ? kcol : idx1[i];
            best1[i] = gt ? v : best1[i];
            best2[i] = nb2;
        }
    }

    // Butterfly top-2 merge across the 16 lanes of each half-wave.
#pragma unroll
    for (int s = 1; s <= 8; s <<= 1) {
#pragma unroll
        for (int i = 0; i < 8; ++i) {
            const float ob1 = __shfl_xor(best1[i], s, 32);
            const int   oi1 = __shfl_xor(idx1[i],  s, 32);
            const float ob2 = __shfl_xor(best2[i], s, 32);
            const bool  take  = ob1 > best1[i];
            const float cross = take ? best1[i] : ob1;      // loser's top-1
            const float n2    = fmaxf(take ? ob2 : best2[i], cross);
            idx1[i]  = take ? oi1 : idx1[i];
            best1[i] = take ? ob1 : best1[i];
            best2[i] = n2;
        }
    }

    // Cross-wave merge via LDS, then compute the per-row loss.
    __shared__ float s_v1[8][16];
    __shared__ int   s_i1[8][16];
    __shared__ float s_v2[8][16];
    if (ln == 0) {  // lane 0 carries rows 0..7, lane 16 carries rows 8..15
#pragma unroll
        for (int i = 0; i < 8; ++i) {
            const int row = half * 8 + i;
            s_v1[w][row] = best1[i];
            s_i1[w][row] = idx1[i];
            s_v2[w][row] = best2[i];
        }
    }
    __syncthreads();

    if (threadIdx.x < 16) {
        const int row = threadIdx.x;
        float a1 = s_v1[0][row]; int i1 = s_i1[0][row]; float a2 = s_v2[0][row];
#pragma unroll
        for (int ww = 1; ww < 8; ++ww) {
            const float b1 = s_v1[ww][row];
            const int   bi = s_i1[ww][row];
            const float b2 = s_v2[ww][row];
            const bool  take  = b1 > a1;
            const float cross = take ? a1 : b1;
            a2 = fmaxf(take ? b2 : a2, cross);
            i1 = take ? bi : i1;
            a1 = take ? b1 : a1;
        }
        const float da0 = 1.0f - a1;   // smallest distance
        const float da1 = 1.0f - a2;   // second smallest
        const int   rb  = r0 + row;
        // Faithful to reference: subset index compared against label VALUE.
        const float an  = (i1 == label[rb]) ? da1 : da0;
        const float l   = ap[rb] - an + SL_MARGIN;
        row_loss[rb] = l > 0.0f ? l : 0.0f;
    }
}

// ---------------------------------------------------------------------------
// Kernel 3: deterministic single-block mean of 4096 row losses.
// ---------------------------------------------------------------------------
__global__ __launch_bounds__(256) void sl_mean_kernel(
    const float* __restrict__ row_loss, float* __restrict__ out) {
    __shared__ float s[256];
    float sum = 0.0f;
    for (int i = threadIdx.x; i < SL_B; i += 256) sum += row_loss[i];
    s[threadIdx.x] = sum;
    __syncthreads();
#pragma unroll
    for (int st = 128; st > 0; st >>= 1) {
        if (threadIdx.x < st) s[threadIdx.x] += s[threadIdx.x + st];
        __syncthreads();
    }
    if (threadIdx.x == 0) out[0] = s[0] * (1.0f / (float)SL_B);
}

// ---------------------------------------------------------------------------
extern "C" void kernel_launch(void* const* d_in, const int* in_sizes, int n_in,
                              void* d_out, int out_size, void* d_ws, size_t ws_size,
                              hipStream_t stream) {
    (void)in_sizes; (void)n_in; (void)out_size; (void)ws_size;
    const float* ftr    = (const float*)d_in[0];
    // d_in[1] = teachor_ftr: unused by the reference computation.
    const float* protos = (const float*)d_in[2];
    const int*   label  = (const int*)d_in[3];
    const int*   lbl    = (const int*)d_in[4];
    float*       out    = (float*)d_out;

    float* ap       = (float*)d_ws;           // 4096 floats
    float* row_loss = ap + SL_B;              // 4096 floats

    // 1) positive distances: 4096 waves (512 blocks x 8 waves)
    sl_ap_kernel<<<SL_B / 8, 256, 0, stream>>>(ftr, protos, label, ap);
    // 2) fused GEMM + top-2 + loss: one block per 16-row tile,
    //    68 KB dynamic LDS for the rotation-swizzled A tile.
    sl_an_topk_kernel<<<SL_B / 16, 256,
                       16 * SL_LDS_STRIDE * sizeof(float), stream>>>(
        ftr, protos, label, lbl, ap, row_loss);
    // 3) mean
    sl_mean_kernel<<<1, 256, 0, stream>>>(row_loss, out);
}